// MHA_49263274885647
// MI455X (gfx1250) — compile-verified
//
#include <hip/hip_runtime.h>
#include <hip/hip_bf16.h>

// MHA fp32 forward for B=4, L=2048, D=1024, H=16, hd=64 on gfx1250 (MI455X).
// Matrix math on v_wmma_f32_16x16x4_f32; tile staging via CDNA5 async
// global->LDS copies (GLOBAL_LOAD_ASYNC_TO_LDS_B128, ASYNCcnt) with double
// buffering. Q A-fragments register-resident; K-parity-split accumulator
// chains for WMMA ILP.
// Workspace layout (needs 128 MB):
//   qp [B,H,L,64] | kp [B,H,L,64] | vp [B,H,L,64] | ctx [B,L,1024]
// d_out layout: out [B*L*1024] floats, then weights [B*2048*2048] floats.

typedef __attribute__((ext_vector_type(2))) float v2f;
typedef __attribute__((ext_vector_type(8))) float v8f;

#define D_MODEL 1024
#define NHEADS 16
#define HEAD_DIM 64
#define BATCH 4
#define SEQ 2048
#define MTOT (BATCH * SEQ) // 8192

static __device__ __forceinline__ v8f wmma_f32(v2f a, v2f b, v8f c) {
  // (neg_a, A, neg_b, B, c_mod, C, reuse_a, reuse_b)
  return __builtin_amdgcn_wmma_f32_16x16x4_f32(false, a, false, b, (short)0, c,
                                               false, false);
}

// Async 16B global->LDS copy (tracked by ASYNCcnt). VDST VGPR holds the
// workgroup-relative LDS byte address (= low 32 bits of the generic pointer
// to __shared__ on AMDGPU). Both addresses 16B-aligned.
static __device__ __forceinline__ void async_f4(const float* g, float* l) {
  const unsigned lds = (unsigned)(size_t)l;
  asm volatile("global_load_async_to_lds_b128 %0, %1, off" ::"v"(lds), "v"(g)
               : "memory");
}
static __device__ __forceinline__ void wait_async0() {
  asm volatile("s_wait_asynccnt 0x0" ::: "memory");
}

// ---------------------------------------------------------------------------
// Generic Y = X @ W^T + bias, X:[8192,1024], W:[N=1024 rows,1024] row-major.
// Block tile 64(M) x 64(N), 8 waves; each wave a 16x32 strip. K streamed in
// 32-wide chunks, async double-buffered. Pitch 36 (144B rows: 16B aligned,
// conflict-free fragment reads). Four parity-split accumulator chains.
// QKV=true scatters output into [B,H,L,hd]; else plain row-major [M,1024].
// ---------------------------------------------------------------------------
#define GP 36 // gemm LDS pitch (floats)

template <bool QKV>
__global__ __launch_bounds__(256) void gemm_xwt_kernel(
    const float* __restrict__ X, const float* __restrict__ W,
    const float* __restrict__ bias, float* __restrict__ Y) {
  __shared__ float Xs[2][64 * GP];
  __shared__ float Ws[2][64 * GP];

  const int tid = threadIdx.x;
  const int lane = tid & 31;
  const int wave = tid >> 5;
  const int wm = wave >> 1; // 0..3 -> M subtile
  const int wn = wave & 1;  // 0..1 -> N strip of 32
  const int m0 = blockIdx.y * 64;
  const int n0 = blockIdx.x * 64;

  v8f acc0e = {}, acc0o = {}, acc1e = {}, acc1o = {};

  auto stage = [&](int buf, int kc) {
#pragma unroll
    for (int it = 0; it < 2; ++it) {
      const int idx = tid + it * 256; // 0..511 float4 slots
      const int r = idx >> 3;
      const int s = idx & 7;
      async_f4(const_cast<float*>(X) + (size_t)(m0 + r) * D_MODEL + kc + s * 4,
               &Xs[buf][r * GP + s * 4]);
      async_f4(const_cast<float*>(W) + (size_t)(n0 + r) * D_MODEL + kc + s * 4,
               &Ws[buf][r * GP + s * 4]);
    }
  };

  stage(0, 0);
  for (int kc = 0; kc < D_MODEL; kc += 32) {
    const int buf = (kc >> 5) & 1;
    wait_async0();
    __syncthreads(); // tile[buf] in LDS; everyone done reading tile[buf^1]
    if (kc + 32 < D_MODEL) stage(buf ^ 1, kc + 32);

#pragma unroll
    for (int kk = 0; kk < 8; ++kk) {
      const int kb = kk * 4 + ((lane >> 4) << 1);
      v2f a, b0, b1;
      const int ar = (wm * 16 + (lane & 15)) * GP + kb;
      a.x = Xs[buf][ar];
      a.y = Xs[buf][ar + 1];
      const int br0 = (wn * 32 + (lane & 15)) * GP + kb;
      const int br1 = (wn * 32 + 16 + (lane & 15)) * GP + kb;
      b0.x = Ws[buf][br0];
      b0.y = Ws[buf][br0 + 1];
      b1.x = Ws[buf][br1];
      b1.y = Ws[buf][br1 + 1];
      if (kk & 1) {
        acc0o = wmma_f32(a, b0, acc0o);
        acc1o = wmma_f32(a, b1, acc1o);
      } else {
        acc0e = wmma_f32(a, b0, acc0e);
        acc1e = wmma_f32(a, b1, acc1e);
      }
    }
    // no trailing barrier: next iteration's leading barrier protects buffers
  }
  const v8f acc0 = acc0e + acc0o;
  const v8f acc1 = acc1e + acc1o;

  // C/D layout: lane -> n = lane&15, rows (lane>>4)*8 + i.
  const int mb = m0 + wm * 16 + ((lane >> 4) << 3);
#pragma unroll
  for (int sub = 0; sub < 2; ++sub) {
    const int n = n0 + wn * 32 + sub * 16 + (lane & 15);
    const float bv = bias[n];
#pragma unroll
    for (int i = 0; i < 8; ++i) {
      const float val = (sub == 0 ? acc0[i] : acc1[i]) + bv;
      const int m = mb + i;
      size_t off;
      if (QKV) {
        // [B,H,L,hd]: b=m/2048, l=m%2048, h=n/64, hd=n%64
        off = (((size_t)((m >> 11) * NHEADS + (n >> 6)) * SEQ) + (m & 2047)) *
                  HEAD_DIM +
              (n & 63);
      } else {
        off = (size_t)m * D_MODEL + n;
      }
      Y[off] = val;
    }
  }
}

// ---------------------------------------------------------------------------
// Fused attention: one workgroup per (b*H+h, 16-query tile).
// LDS: sQ 16x68 | sK 2 x 128x68 (K then V, double buffered, async-staged)
//      | sS 16x2050 scores->probs | sC 2x16x64 ctx partials
// ---------------------------------------------------------------------------
#define QP_ 68   // sQ pitch
#define KP_ 68   // sK pitch (rows 272B -> 16B aligned)
#define SP_ 2050 // sS pitch
#define ATTN_LDS_FLOATS (16 * QP_ + 2 * 128 * KP_ + 16 * SP_ + 2048)
#define ATTN_LDS_BYTES (ATTN_LDS_FLOATS * 4) // 213376 B < 320 KB WGP LDS

__global__ __launch_bounds__(256) void attn_kernel(
    const float* __restrict__ qp, const float* __restrict__ kp,
    const float* __restrict__ vp, float* __restrict__ ctx,
    float* __restrict__ wmean /* pre-zeroed [B,2048,2048] */) {
  extern __shared__ float smem[];
  float* sQ = smem;             // 16 x 68
  float* sK0 = sQ + 16 * QP_;   // 128 x 68 buffer 0
  float* sK1 = sK0 + 128 * KP_; // 128 x 68 buffer 1
  float* sS = sK1 + 128 * KP_;  // 16 x 2050
  float* sC = sS + 16 * SP_;    // 2 x 16 x 64

  const int q0 = blockIdx.x * 16;
  const int bh = blockIdx.y; // b*16 + h (matches [B,H,...] flattening)
  const int b = bh >> 4;
  const int h = bh & 15;
  const int tid = threadIdx.x;
  const int lane = tid & 31;
  const int wave = tid >> 5;

  const float* Qg = qp + (size_t)bh * SEQ * HEAD_DIM;
  const float* Kg = kp + (size_t)bh * SEQ * HEAD_DIM;
  const float* Vg = vp + (size_t)bh * SEQ * HEAD_DIM;

  // stage a 128x64 row chunk of src into an sK buffer (2048 float4, 8/thread)
  auto stageKV = [&](const float* src, float* dst, int chunk) {
#pragma unroll
    for (int it = 0; it < 8; ++it) {
      const int idx = tid + it * 256; // float4 index 0..2047
      const int r = idx >> 4;
      const int s = idx & 15;
      async_f4(const_cast<float*>(src) + (size_t)(chunk * 128 + r) * HEAD_DIM +
                   s * 4,
               &dst[r * KP_ + s * 4]);
    }
  };

  // ---- Phase 1: async-stage Q tile (16x64 = 256 float4, 1/thread) + K0 ---
  {
    const int r = tid >> 4;
    const int s = tid & 15;
    async_f4(const_cast<float*>(Qg) + (size_t)(q0 + r) * HEAD_DIM + s * 4,
             &sQ[r * QP_ + s * 4]);
  }
  stageKV(Kg, sK0, 0);
  wait_async0();
  __syncthreads(); // Q tile and K chunk 0 resident

  // Preload the 16 Q A-fragments (invariant across all K chunks): 32 VGPRs.
  v2f aq[16];
#pragma unroll
  for (int kk = 0; kk < 16; ++kk) {
    const int ar = (lane & 15) * QP_ + kk * 4 + ((lane >> 4) << 1);
    aq[kk].x = sQ[ar];
    aq[kk].y = sQ[ar + 1];
  }

  // ---- Phase 2: raw scores S = Q @ K^T (scale folded into softmax) -------
  for (int chunk = 0; chunk < 16; ++chunk) {
    float* cur = (chunk & 1) ? sK1 : sK0;
    float* nxt = (chunk & 1) ? sK0 : sK1;
    if (chunk > 0) { // chunk 0 already waited above
      wait_async0();
      __syncthreads();
    }
    if (chunk < 15) stageKV(Kg, nxt, chunk + 1);

    v8f accE = {}, accO = {};
#pragma unroll
    for (int kk = 0; kk < 16; ++kk) { // K = 64 / 4
      const int kb = kk * 4 + ((lane >> 4) << 1);
      v2f bf;
      const int br = (wave * 16 + (lane & 15)) * KP_ + kb;
      bf.x = cur[br];
      bf.y = cur[br + 1];
      if (kk & 1)
        accO = wmma_f32(aq[kk], bf, accO);
      else
        accE = wmma_f32(aq[kk], bf, accE);
    }
    const v8f acc = accE + accO;
    const int col = chunk * 128 + wave * 16 + (lane & 15);
    const int mb = (lane >> 4) << 3;
#pragma unroll
    for (int i = 0; i < 8; ++i) sS[(mb + i) * SP_ + col] = acc[i];
  }
  __syncthreads(); // all score columns visible

  // ---- Phase 3: row softmax of s/8 (wave32 shuffle reductions) -----------
#pragma unroll
  for (int rr = 0; rr < 2; ++rr) {
    const int r = wave * 2 + rr;
    float mx = -INFINITY;
    for (int c = lane; c < SEQ; c += 32) mx = fmaxf(mx, sS[r * SP_ + c]);
    for (int off = 16; off; off >>= 1) mx = fmaxf(mx, __shfl_xor(mx, off, 32));
    float sum = 0.f;
    for (int c = lane; c < SEQ; c += 32) {
      const float e = __expf((sS[r * SP_ + c] - mx) * 0.125f); // 1/sqrt(64)
      sS[r * SP_ + c] = e;
      sum += e;
    }
    for (int off = 16; off; off >>= 1) sum += __shfl_xor(sum, off, 32);
    const float inv = 1.0f / sum;
    for (int c = lane; c < SEQ; c += 32) sS[r * SP_ + c] *= inv;
  }
  __syncthreads();

  // kick off V chunk 0 now; it overlaps the weights atomics below
  stageKV(Vg, sK0, 0);

  // ---- Phase 4: head-mean weights via fp32 global atomics ----------------
  {
    float* wout = wmean + ((size_t)b * SEQ + q0) * SEQ;
    for (int idx = tid; idx < 16 * SEQ; idx += 256) {
      const int r = idx >> 11, c = idx & 2047;
      atomicAdd(&wout[(size_t)r * SEQ + c], sS[r * SP_ + c] * (1.0f / NHEADS));
    }
  }

  // ---- Phase 5: ctx = P @ V; waves split 4 n-tiles x 2 K-halves ----------
  v8f caccE = {}, caccO = {};
  const int nt = wave & 3;  // 16-wide column tile of hd
  const int kh = wave >> 2; // K half within each 128-chunk
  for (int chunk = 0; chunk < 16; ++chunk) {
    float* cur = (chunk & 1) ? sK1 : sK0;
    float* nxt = (chunk & 1) ? sK0 : sK1;
    wait_async0();
    __syncthreads();
    if (chunk < 15) stageKV(Vg, nxt, chunk + 1);

#pragma unroll
    for (int kk = 0; kk < 16; ++kk) { // 64 K-values per half / 4
      const int kl = kh * 64 + kk * 4 + ((lane >> 4) << 1);
      const int kg = chunk * 128 + kl;
      v2f a, bf;
      const int ar = (lane & 15) * SP_ + kg;
      a.x = sS[ar];
      a.y = sS[ar + 1];
      bf.x = cur[kl * KP_ + nt * 16 + (lane & 15)];
      bf.y = cur[(kl + 1) * KP_ + nt * 16 + (lane & 15)];
      if (kk & 1)
        caccO = wmma_f32(a, bf, caccO);
      else
        caccE = wmma_f32(a, bf, caccE);
    }
  }
  const v8f cacc = caccE + caccO;
  // store partials, reduce the two K-halves, write ctx in [B,L,H*hd]
  {
    const int mb = (lane >> 4) << 3;
#pragma unroll
    for (int i = 0; i < 8; ++i)
      sC[kh * 1024 + (mb + i) * 64 + nt * 16 + (lane & 15)] = cacc[i];
  }
  __syncthreads();
  {
    float* cg = ctx + ((size_t)b * SEQ + q0) * D_MODEL + h * HEAD_DIM;
    const int idx = tid * 4; // 1024 floats / 256 threads = one float4 each
    const float4 p0 = *(const float4*)(sC + idx);
    const float4 p1 = *(const float4*)(sC + 1024 + idx);
    float4 o;
    o.x = p0.x + p1.x;
    o.y = p0.y + p1.y;
    o.z = p0.z + p1.z;
    o.w = p0.w + p1.w;
    const int m = idx >> 6, c = idx & 63;
    *(float4*)(cg + (size_t)m * D_MODEL + c) = o;
  }
}

__global__ void zero_kernel(float* __restrict__ p, size_t n) {
  size_t i = (size_t)blockIdx.x * blockDim.x + threadIdx.x;
  const size_t stride = (size_t)gridDim.x * blockDim.x;
  for (; i < n; i += stride) p[i] = 0.0f;
}

extern "C" void kernel_launch(void* const* d_in, const int* in_sizes, int n_in,
                              void* d_out, int out_size, void* d_ws,
                              size_t ws_size, hipStream_t stream) {
  (void)in_sizes; (void)n_in; (void)out_size; (void)ws_size;
  const float* q = (const float*)d_in[0];
  const float* k = (const float*)d_in[1];
  const float* v = (const float*)d_in[2];
  const float* in_w = (const float*)d_in[3];  // [3*1024, 1024]
  const float* in_b = (const float*)d_in[4];  // [3*1024]
  const float* out_w = (const float*)d_in[5]; // [1024, 1024]
  const float* out_b = (const float*)d_in[6]; // [1024]

  float* out = (float*)d_out;                  // [8192*1024]
  float* wmean = out + (size_t)MTOT * D_MODEL; // [4*2048*2048]

  const size_t QKV_N = (size_t)MTOT * D_MODEL; // 8,388,608 floats
  float* qp = (float*)d_ws;
  float* kp = qp + QKV_N;
  float* vp = kp + QKV_N;
  float* ctx = vp + QKV_N;

  const dim3 gemm_grid(D_MODEL / 64, MTOT / 64); // (16, 128)
  gemm_xwt_kernel<true><<<gemm_grid, 256, 0, stream>>>(q, in_w, in_b, qp);
  gemm_xwt_kernel<true><<<gemm_grid, 256, 0, stream>>>(
      k, in_w + (size_t)D_MODEL * D_MODEL, in_b + D_MODEL, kp);
  gemm_xwt_kernel<true><<<gemm_grid, 256, 0, stream>>>(
      v, in_w + (size_t)2 * D_MODEL * D_MODEL, in_b + 2 * D_MODEL, vp);

  zero_kernel<<<4096, 256, 0, stream>>>(wmean, (size_t)BATCH * SEQ * SEQ);

  static_assert(ATTN_LDS_BYTES < 320 * 1024, "fits WGP LDS");
  hipFuncSetAttribute(reinterpret_cast<const void*>(&attn_kernel),
                      hipFuncAttributeMaxDynamicSharedMemorySize,
                      ATTN_LDS_BYTES);
  attn_kernel<<<dim3(SEQ / 16, BATCH * NHEADS), 256, ATTN_LDS_BYTES, stream>>>(
      qp, kp, vp, ctx, wmean);

  gemm_xwt_kernel<false><<<gemm_grid, 256, 0, stream>>>(ctx, out_w, out_b, out);
}